// SDF_Loss_72670846649071
// MI455X (gfx1250) — compile-verified
//
#include <hip/hip_runtime.h>
#include <hip/hip_bf16.h>

#define G 32
#define G3 (G * G * G)
#define RADIUS 0.3f
#define SCALE_FACTOR 0.2f
#define SDF_WEIGHT 1000.0f
#define PP 2               // num_people, fixed by reference setup_inputs()
#define MT 4               // M-tiles (of 16 grid points) per wave
#define WAVES_PER_BLOCK 8  // 256 threads, wave32
#define BPP ((G3 / 16) / (WAVES_PER_BLOCK * MT))  // blocks per person = 64

typedef __attribute__((ext_vector_type(2))) float v2f;
typedef __attribute__((ext_vector_type(8))) float v8f;

// ---------------------------------------------------------------- init losses
__global__ void k_init(float* __restrict__ losses, int F) {
    int t = blockIdx.x * blockDim.x + threadIdx.x;
    if (t < F) losses[t] = 0.0f;
}

// ------------------------------------------------- per-person bbox -> center/scale
__global__ void k_bounds(const float* __restrict__ verts,
                         const float* __restrict__ camt,
                         float* __restrict__ cs, int V) {
    int person = blockIdx.x;
    const float* vp = verts + (size_t)person * V * 3;
    float tx = camt[person * 3 + 0];
    float ty = camt[person * 3 + 1];
    float tz = camt[person * 3 + 2];
    float mnx = 1e30f, mny = 1e30f, mnz = 1e30f;
    float mxx = -1e30f, mxy = -1e30f, mxz = -1e30f;
    for (int v = threadIdx.x; v < V; v += blockDim.x) {
        float x = vp[3 * v + 0] + tx;
        float y = vp[3 * v + 1] + ty;
        float z = vp[3 * v + 2] + tz;
        mnx = fminf(mnx, x); mxx = fmaxf(mxx, x);
        mny = fminf(mny, y); mxy = fmaxf(mxy, y);
        mnz = fminf(mnz, z); mxz = fmaxf(mxz, z);
    }
    __shared__ float smn[3][256];
    __shared__ float smx[3][256];
    int t = threadIdx.x;
    smn[0][t] = mnx; smn[1][t] = mny; smn[2][t] = mnz;
    smx[0][t] = mxx; smx[1][t] = mxy; smx[2][t] = mxz;
    __syncthreads();
    for (int s = 128; s > 0; s >>= 1) {
        if (t < s) {
            for (int a = 0; a < 3; ++a) {
                smn[a][t] = fminf(smn[a][t], smn[a][t + s]);
                smx[a][t] = fmaxf(smx[a][t], smx[a][t + s]);
            }
        }
        __syncthreads();
    }
    if (t == 0) {
        float ex = smx[0][0] - smn[0][0];
        float ey = smx[1][0] - smn[1][0];
        float ez = smx[2][0] - smn[2][0];
        float ext = fmaxf(ex, fmaxf(ey, ez));
        cs[person * 4 + 0] = 0.5f * (smn[0][0] + smx[0][0]);
        cs[person * 4 + 1] = 0.5f * (smn[1][0] + smx[1][0]);
        cs[person * 4 + 2] = 0.5f * (smn[2][0] + smx[2][0]);
        cs[person * 4 + 3] = (1.0f + SCALE_FACTOR) * 0.5f * ext;
    }
}

// ---------------------------------------- build padded (vx,vy,vz,|v|^2) B table
__global__ void k_vb(const float* __restrict__ verts,
                     const float* __restrict__ camt,
                     float4* __restrict__ vb, int V, int nslot, int nPers) {
    int idx = blockIdx.x * blockDim.x + threadIdx.x;
    if (idx >= nPers * nslot) return;
    int person = idx / nslot;
    int slot = idx - person * nslot;
    float4 o;
    if (slot < V) {
        const float* vp = verts + ((size_t)person * V + slot) * 3;
        float x = vp[0] + camt[person * 3 + 0];
        float y = vp[1] + camt[person * 3 + 1];
        float z = vp[2] + camt[person * 3 + 2];
        o = make_float4(x, y, z, x * x + y * y + z * z);
    } else {
        o = make_float4(0.0f, 0.0f, 0.0f, 1e10f);  // never wins the min
    }
    vb[idx] = o;
}

// --------------------------------------------------------- phi via f32 WMMA K=4
// d^2(m,n) = |p_m|^2 + ( |v_n|^2 - 2 p_m . v_n )
// A row m = (-2px, -2py, -2pz, 1) ; B col n = (vx, vy, vz, |v|^2)
// nvtp: padded (even) number of 16-vertex tiles
__global__ void __launch_bounds__(256) k_phi(const float4* __restrict__ vb,
                                             const float* __restrict__ cs,
                                             float* __restrict__ phi,
                                             int nvtp, int bpp) {
    int person = blockIdx.x / bpp;
    int blk = blockIdx.x - person * bpp;
    int lane = threadIdx.x & 31;
    int wave = threadIdx.x >> 5;
    int tileBase = (blk * WAVES_PER_BLOCK + wave) * MT;  // M-tile index

    float cx = cs[person * 4 + 0];
    float cy = cs[person * 4 + 1];
    float cz = cs[person * 4 + 2];
    float s  = cs[person * 4 + 3];
    const float4* vbp = vb + (size_t)person * nvtp * 16;
    float* php = phi + (size_t)person * G3;

    bool hi = (lane >= 16);
    int ml = lane & 15;

    v2f A[MT];
    float pm2[MT];
    v8f dmin[MT];
#pragma unroll
    for (int t = 0; t < MT; ++t) {
        int mg = (tileBase + t) * 16 + ml;  // global grid index
        int ix = mg >> 10;
        int iy = (mg >> 5) & (G - 1);
        int iz = mg & (G - 1);
        float px = (-1.0f + (float)ix * (2.0f / 31.0f)) * s + cx;
        float py = (-1.0f + (float)iy * (2.0f / 31.0f)) * s + cy;
        float pz = (-1.0f + (float)iz * (2.0f / 31.0f)) * s + cz;
        // A VGPR0: K=0 (lanes 0-15) / K=2 (lanes 16-31); VGPR1: K=1 / K=3
        A[t].x = hi ? (-2.0f * pz) : (-2.0f * px);
        A[t].y = hi ? 1.0f         : (-2.0f * py);
        pm2[t] = px * px + py * py + pz * pz;
#pragma unroll
        for (int r = 0; r < 8; ++r) dmin[t][r] = 1e10f;  // matches reference init
    }

    v8f zero = {};
    // unroll-by-2 over vertex tiles: two live WMMA results feed one v_min3
    for (int vt = 0; vt < nvtp; vt += 2) {
        float4 v4a = vbp[vt * 16 + ml];
        float4 v4b = vbp[(vt + 1) * 16 + ml];
        // B VGPR0: K=0 / K=2 ; VGPR1: K=1 / K=3
        v2f Ba, Bb;
        Ba.x = hi ? v4a.z : v4a.x;
        Ba.y = hi ? v4a.w : v4a.y;
        Bb.x = hi ? v4b.z : v4b.x;
        Bb.y = hi ? v4b.w : v4b.y;
#pragma unroll
        for (int t = 0; t < MT; ++t) {
            v8f d0 = __builtin_amdgcn_wmma_f32_16x16x4_f32(
                false, A[t], false, Ba, (short)0, zero, false, false);
            v8f d1 = __builtin_amdgcn_wmma_f32_16x16x4_f32(
                false, A[t], false, Bb, (short)0, zero, false, false);
#pragma unroll
            for (int r = 0; r < 8; ++r) {
                float dm = dmin[t][r];
                // 3-input min, no canonicalize traffic (gfx12 "_num" naming)
                asm("v_min3_num_f32 %0, %1, %2, %3"
                    : "=v"(dm)
                    : "v"(dm), "v"(d0[r]), "v"(d1[r]));
                dmin[t][r] = dm;
            }
        }
    }

#pragma unroll
    for (int t = 0; t < MT; ++t) {
#pragma unroll
        for (int r = 0; r < 8; ++r) {
            // min across the 16 lanes of each half (row m's columns)
            float red = dmin[t][r];
            red = fminf(red, __shfl_xor(red, 1));
            red = fminf(red, __shfl_xor(red, 2));
            red = fminf(red, __shfl_xor(red, 4));
            red = fminf(red, __shfl_xor(red, 8));
            int m = r + (hi ? 8 : 0);          // D layout: M = r + 8*(lane>=16)
            float p2 = __shfl(pm2[t], m);      // |p_m|^2 lives in lane m
            float d2 = fmaxf(red + p2, 0.0f);
            float dd = sqrtf(d2);
            float ph = fmaxf(RADIUS - dd, 0.0f);
            if (ml == 0) php[(tileBase + t) * 16 + m] = ph;  // lanes 0 and 16 write
        }
    }
}

// ------------------------------------------- trilinear sample of phi, pair sum
__global__ void k_sample(const float* __restrict__ verts,
                         const float* __restrict__ camt,
                         const float* __restrict__ cs,
                         const float* __restrict__ phi,
                         float* __restrict__ losses, int V, int P) {
    int pairsPerFrame = P * (P - 1);
    int frame = blockIdx.x / pairsPerFrame;
    int rem = blockIdx.x - frame * pairsPerFrame;
    int i = rem / (P - 1);
    int jj = rem - i * (P - 1);
    int j = jj + (jj >= i ? 1 : 0);
    int pi = frame * P + i;  // phi owner
    int pj = frame * P + j;  // vertices sampled into phi_i

    float cxi = cs[pi * 4 + 0];
    float cyi = cs[pi * 4 + 1];
    float czi = cs[pi * 4 + 2];
    float inv = 1.0f / cs[pi * 4 + 3];
    const float* ph = phi + (size_t)pi * G3;
    const float* vp = verts + (size_t)pj * V * 3;
    float tx = camt[pj * 3 + 0];
    float ty = camt[pj * 3 + 1];
    float tz = camt[pj * 3 + 2];

    float acc = 0.0f;
    for (int v = threadIdx.x; v < V; v += blockDim.x) {
        float lx = (vp[3 * v + 0] + tx - cxi) * inv;
        float ly = (vp[3 * v + 1] + ty - cyi) * inv;
        float lz = (vp[3 * v + 2] + tz - czi) * inv;
        float gx = fminf(fmaxf((lx + 1.0f) * 0.5f * (G - 1), 0.0f), (float)(G - 1));
        float gy = fminf(fmaxf((ly + 1.0f) * 0.5f * (G - 1), 0.0f), (float)(G - 1));
        float gz = fminf(fmaxf((lz + 1.0f) * 0.5f * (G - 1), 0.0f), (float)(G - 1));
        int x0 = min((int)gx, G - 2);
        int y0 = min((int)gy, G - 2);
        int z0 = min((int)gz, G - 2);
        float fx = gx - (float)x0;
        float fy = gy - (float)y0;
        float fz = gz - (float)z0;
        int x1 = x0 + 1, y1 = y0 + 1, z1 = z0 + 1;
        const float c000 = ph[(x0 * G + y0) * G + z0];
        const float c100 = ph[(x1 * G + y0) * G + z0];
        const float c010 = ph[(x0 * G + y1) * G + z0];
        const float c110 = ph[(x1 * G + y1) * G + z0];
        const float c001 = ph[(x0 * G + y0) * G + z1];
        const float c101 = ph[(x1 * G + y0) * G + z1];
        const float c011 = ph[(x0 * G + y1) * G + z1];
        const float c111 = ph[(x1 * G + y1) * G + z1];
        float c00 = c000 * (1.0f - fx) + c100 * fx;
        float c10 = c010 * (1.0f - fx) + c110 * fx;
        float c01 = c001 * (1.0f - fx) + c101 * fx;
        float c11 = c011 * (1.0f - fx) + c111 * fx;
        float c0 = c00 * (1.0f - fy) + c10 * fy;
        float c1 = c01 * (1.0f - fy) + c11 * fy;
        float sres = c0 * (1.0f - fz) + c1 * fz;
        if (sres > 0.0f) acc += sres;  // THRESH = 0
    }
    __shared__ float sred[256];
    sred[threadIdx.x] = acc;
    __syncthreads();
    for (int s = 128; s > 0; s >>= 1) {
        if (threadIdx.x < s) sred[threadIdx.x] += sred[threadIdx.x + s];
        __syncthreads();
    }
    if (threadIdx.x == 0)
        atomicAdd(&losses[frame], sred[0] / (float)(P * P));
}

// ----------------------------------------------------- valid-weighted mean * W
__global__ void k_final(const float* __restrict__ losses,
                        const int* __restrict__ valid,
                        float* __restrict__ out, int F, int P) {
    if (threadIdx.x == 0 && blockIdx.x == 0) {
        float sl = 0.0f, sv = 0.0f;
        for (int f = 0; f < F; ++f) {
            float vf = (float)valid[f * P];
            sl += losses[f] * vf;
            sv += vf;
        }
        out[0] = sl / sv * SDF_WEIGHT;
    }
}

extern "C" void kernel_launch(void* const* d_in, const int* in_sizes, int n_in,
                              void* d_out, int out_size, void* d_ws, size_t ws_size,
                              hipStream_t stream) {
    const float* verts = (const float*)d_in[0];
    const float* camt  = (const float*)d_in[1];
    const int*   valid = (const int*)d_in[2];

    int n = in_sizes[2];            // total persons = bs*frame_length*P = 32
    int V = in_sizes[0] / (n * 3);  // 6890
    int P = PP;
    int F = n / P;                  // 16 frames
    int nvt16 = (V + 15) / 16;      // 431 vertex tiles
    int nvtp = (nvt16 + 1) & ~1;    // padded to even (432) for unroll-by-2
    int nslot = nvtp * 16;

    float* ws = (float*)d_ws;
    float4* vb   = (float4*)ws;                          // n*nslot float4
    float* phi    = ws + (size_t)n * nslot * 4;          // n*G3
    float* cs     = phi + (size_t)n * G3;                // n*4
    float* losses = cs + (size_t)n * 4;                  // F
    float* out    = (float*)d_out;

    k_init<<<1, 64, 0, stream>>>(losses, F);
    k_bounds<<<n, 256, 0, stream>>>(verts, camt, cs, V);
    int tot = n * nslot;
    k_vb<<<(tot + 255) / 256, 256, 0, stream>>>(verts, camt, vb, V, nslot, n);
    k_phi<<<n * BPP, 256, 0, stream>>>(vb, cs, phi, nvtp, BPP);
    k_sample<<<F * P * (P - 1), 256, 0, stream>>>(verts, camt, cs, phi, losses, V, P);
    k_final<<<1, 32, 0, stream>>>(losses, valid, out, F, P);
}